// MySageConvHistory_39522289058163
// MI455X (gfx1250) — compile-verified
//
#include <hip/hip_runtime.h>
#include <hip/hip_bf16.h>

typedef __attribute__((ext_vector_type(2))) float v2f;
typedef __attribute__((ext_vector_type(8))) float v8f;

#define FEAT 256

// ---------------------------------------------------------------------------
// Kernel 1: segment-mean aggregation + history-cache select.
// One 64-lane group per node, float4 per lane (64*4 = 256 features).
// Writes `his` directly into the second output slot of d_out.
// ---------------------------------------------------------------------------
__global__ __launch_bounds__(256) void sage_aggregate(
    const float* __restrict__ x, const int* __restrict__ ptr,
    const int* __restrict__ idx, const int* __restrict__ hmap,
    const float* __restrict__ hbuf, float* __restrict__ his,
    int num_node, int hist_size) {
  const int group = threadIdx.x >> 6;   // 4 nodes per 256-thread block
  const int t     = threadIdx.x & 63;
  const int node  = blockIdx.x * 4 + group;
  if (node >= num_node) return;
  const int f = t * 4;

  float4 r;
  const int hm = hmap[node];            // uniform per wave
  if (hm >= 0) {
    int h = hm < hist_size ? hm : hist_size - 1;
    r = *(const float4*)(hbuf + (size_t)h * FEAT + f);
  } else {
    const int s = ptr[node], e = ptr[node + 1];
    float4 acc = make_float4(0.f, 0.f, 0.f, 0.f);
    for (int j = s; j < e; ++j) {
      const int row = idx[j];
      float4 v = *(const float4*)(x + (size_t)row * FEAT + f);
      acc.x += v.x; acc.y += v.y; acc.z += v.z; acc.w += v.w;
    }
    float d = (float)(e - s);
    d = d > 1.f ? d : 1.f;              // jnp.maximum(counts, 1.0)
    r.x = acc.x / d; r.y = acc.y / d; r.z = acc.z / d; r.w = acc.w / d;
  }
  *(float4*)(his + (size_t)node * FEAT + f) = r;
}

// ---------------------------------------------------------------------------
// Kernel 2: fused GEMM  out[m,h] = b_l[h] + sum_k his[m,k]W_l[h,k] + x[m,k]W_r[h,k]
// Single M x 256 x 512 GEMM (K-concat) via V_WMMA_F32_16X16X4_F32.
// Wave tile: 32 M x 64 N  -> 8 accumulator tiles; two A frags share each B frag
// (0.75 loads per WMMA). Block = 8 waves = 64 M x 256 N.
// ---------------------------------------------------------------------------
__global__ __launch_bounds__(256) void sage_gemm(
    const float* __restrict__ his,   // [M, 256]
    const float* __restrict__ x,     // [>=M, 256]
    const float* __restrict__ Wl,    // [256, 256] row-major (HID x IN)
    const float* __restrict__ Wr,    // [256, 256]
    const float* __restrict__ bl,    // [256]
    float* __restrict__ out,         // [M, 256]
    int M) {
  const int lane   = threadIdx.x & 31;
  const int wave   = threadIdx.x >> 5;        // 0..7
  const int mbase  = blockIdx.x * 64 + (wave >> 2) * 32;  // 32-row wave tile
  const int nstrip = (wave & 3) * 64;
  if (mbase >= M) return;                     // uniform per wave, before any WMMA

  const int lrow = lane & 15;                 // M-index for A frag, N-index for B frag
  const int kk   = (lane >> 4) * 2;           // K sub-offset within fragment

  // Accumulators initialized with bias (every C/D VGPR of a tile holds same N).
  v8f acc[2][4];
#pragma unroll
  for (int t = 0; t < 4; ++t) {
    const float b = bl[nstrip + t * 16 + lrow];
#pragma unroll
    for (int j = 0; j < 8; ++j) { acc[0][t][j] = b; acc[1][t][j] = b; }
  }

  // Clamp A-row indices: keeps EXEC all-ones (WMMA requirement).
  int ar0 = mbase + lrow;       if (ar0 >= M) ar0 = M - 1;
  int ar1 = mbase + 16 + lrow;  if (ar1 >= M) ar1 = M - 1;
  const float* __restrict__ aH0 = his + (size_t)ar0 * FEAT;
  const float* __restrict__ aH1 = his + (size_t)ar1 * FEAT;
  const float* __restrict__ aX0 = x   + (size_t)ar0 * FEAT;
  const float* __restrict__ aX1 = x   + (size_t)ar1 * FEAT;

  // Two K-halves: his * Wl then x * Wr (virtual K = 512).
  for (int half = 0; half < 2; ++half) {
    const float* __restrict__ A0 = half ? aX0 : aH0;
    const float* __restrict__ A1 = half ? aX1 : aH1;
    const float* __restrict__ W  = half ? Wr  : Wl;
#pragma unroll 2
    for (int k0 = 0; k0 < FEAT; k0 += 4) {
      // A fragments: lane holds A[m][k0+kk .. k0+kk+1] (contiguous float2)
      v2f a0 = *(const v2f*)(A0 + k0 + kk);
      v2f a1 = *(const v2f*)(A1 + k0 + kk);
#pragma unroll
      for (int t = 0; t < 4; ++t) {
        const int n = nstrip + t * 16 + lrow;
        // B fragment: lane holds W[n][k0+kk .. k0+kk+1] (contiguous float2)
        v2f b = *(const v2f*)(W + (size_t)n * FEAT + k0 + kk);
        acc[0][t] = __builtin_amdgcn_wmma_f32_16x16x4_f32(
            false, a0, false, b, (short)0, acc[0][t], false, false);
        acc[1][t] = __builtin_amdgcn_wmma_f32_16x16x4_f32(
            false, a1, false, b, (short)0, acc[1][t], false, false);
      }
    }
  }

  // Store: D VGPR j -> row mbase + s*16 + j + (lane>=16)*8, col nstrip+t*16+(lane&15)
#pragma unroll
  for (int s = 0; s < 2; ++s) {
#pragma unroll
    for (int t = 0; t < 4; ++t) {
      const int n = nstrip + t * 16 + lrow;
#pragma unroll
      for (int j = 0; j < 8; ++j) {
        const int m = mbase + s * 16 + j + (lane >> 4) * 8;
        if (m < M) out[(size_t)m * FEAT + n] = acc[s][t][j];
      }
    }
  }
}

// ---------------------------------------------------------------------------
// Inputs (setup_inputs order):
//  0 x [500000,256] f32      1 ptr [100001] i32       2 idx [1.6M] i32
//  3 history_map [100000]    4 history_buffer [20000,256] f32
//  5 history_size (scalar)   6 num_node (scalar)
//  7 W_l [256,256] f32       8 b_l [256] f32          9 W_r [256,256] f32
// d_out = [out (M*256) | his (M*256)] f32
// ---------------------------------------------------------------------------
extern "C" void kernel_launch(void* const* d_in, const int* in_sizes, int n_in,
                              void* d_out, int out_size, void* d_ws, size_t ws_size,
                              hipStream_t stream) {
  const float* x    = (const float*)d_in[0];
  const int*   ptr  = (const int*)d_in[1];
  const int*   idx  = (const int*)d_in[2];
  const int*   hmap = (const int*)d_in[3];
  const float* hbuf = (const float*)d_in[4];
  const float* Wl   = (const float*)d_in[7];
  const float* bl   = (const float*)d_in[8];
  const float* Wr   = (const float*)d_in[9];

  const int num_node  = in_sizes[3];          // len(history_map)
  const int hist_size = in_sizes[4] / FEAT;   // rows of history_buffer

  float* out = (float*)d_out;
  float* his = out + (size_t)num_node * FEAT; // second tuple output, also GEMM input

  const int agg_blocks = (num_node + 3) / 4;
  sage_aggregate<<<agg_blocks, 256, 0, stream>>>(x, ptr, idx, hmap, hbuf, his,
                                                 num_node, hist_size);

  const int gemm_blocks = (num_node + 63) / 64;
  sage_gemm<<<gemm_blocks, 256, 0, stream>>>(his, x, Wl, Wr, bl, out, num_node);
}